// MemN2N_12970801234095
// MI455X (gfx1250) — compile-verified
//
#include <hip/hip_runtime.h>

#define VV 50000
#define DD 128
#define SS 50
#define QQ 20
#define MM 200
#define BB 64

typedef __attribute__((ext_vector_type(2))) float v2f;
typedef __attribute__((ext_vector_type(8))) float v8f;

// ---------------------------------------------------------------------------
// emb[t][b*M+m][d] = sum_s T_t[story[b,m,s], d] * PE_sent(s, d)
// grid: (B*M, 4), block: 128
// ---------------------------------------------------------------------------
__global__ void memn2n_embed(const int* __restrict__ story,
                             const float* __restrict__ T0,
                             const float* __restrict__ T1,
                             const float* __restrict__ T2,
                             const float* __restrict__ T3,
                             float* __restrict__ emb) {
  const int bm = blockIdx.x;            // 0 .. B*M-1
  const int t  = blockIdx.y;            // table id 0..3
  const int d  = threadIdx.x;           // 0..127
  const float* tab = (t == 0) ? T0 : (t == 1) ? T1 : (t == 2) ? T2 : T3;
  const float kf = (float)(d + 1) * (1.0f / (float)DD);
  const int* srow = story + bm * SS;
  float acc = 0.0f;
#pragma unroll 5
  for (int s = 0; s < SS; ++s) {
    const float jf = (float)(s + 1) * (1.0f / (float)SS);
    const float pe = 1.0f - jf - kf * (1.0f - 2.0f * jf);
    const int idx = srow[s];            // uniform per block -> scalar load
    acc += tab[idx * DD + d] * pe;      // coalesced 512B row gather
  }
  emb[((size_t)t * (BB * MM) + bm) * DD + d] = acc;
}

// ---------------------------------------------------------------------------
// u0[b][d] = sum_q B_emb[question[b,q], d] * PE_query(q, d)
// grid: B, block: 128
// ---------------------------------------------------------------------------
__global__ void memn2n_query(const int* __restrict__ question,
                             const float* __restrict__ Bemb,
                             float* __restrict__ u) {
  const int b = blockIdx.x;
  const int d = threadIdx.x;
  const float kf = (float)(d + 1) * (1.0f / (float)DD);
  const int* qrow = question + b * QQ;
  float acc = 0.0f;
#pragma unroll
  for (int q = 0; q < QQ; ++q) {
    const float jf = (float)(q + 1) * (1.0f / (float)QQ);
    const float pe = 1.0f - jf - kf * (1.0f - 2.0f * jf);
    acc += Bemb[qrow[q] * DD + d] * pe;
  }
  u[b * DD + d] = acc;
}

// ---------------------------------------------------------------------------
// 3 fused hops per batch element. grid: B, block: 256.
// A_tabs[k] = emb[k], C_tabs[k] = emb[k+1] (adjacent weight tying).
// ---------------------------------------------------------------------------
__global__ void memn2n_hops(const int* __restrict__ story,
                            const float* __restrict__ TA,
                            const float* __restrict__ TC,
                            const float* __restrict__ emb,
                            float* __restrict__ u) {
  const int b = blockIdx.x;
  const int tid = threadIdx.x;
  __shared__ float u_s[DD];
  __shared__ float sc[MM];
  __shared__ int   np[MM];
  __shared__ float red[2];

  // non-pad mask: any(story[b,m,:] != 0)
  for (int m = tid; m < MM; m += blockDim.x) {
    const int* srow = story + (b * MM + m) * SS;
    int any = 0;
    for (int s = 0; s < SS; ++s) any |= srow[s];
    np[m] = (any != 0);
  }
  if (tid < DD) u_s[tid] = u[b * DD + tid];
  __syncthreads();

  for (int k = 0; k < 3; ++k) {
    const float* ea = emb + ((size_t)k       * (BB * MM) + b * MM) * DD;
    const float* ec = emb + ((size_t)(k + 1) * (BB * MM) + b * MM) * DD;
    // scores[m] = dot(ea[m] + TA[m], u)
    if (tid < MM) {
      const float* row = ea + tid * DD;
      const float* ta  = TA + tid * DD;
      float s = 0.0f;
      for (int d0 = 0; d0 < DD; ++d0) s += (row[d0] + ta[d0]) * u_s[d0];
      sc[tid] = np[tid] ? s : -1e9f;
    }
    __syncthreads();
    if (tid == 0) {
      float mx = -1e30f;
      for (int m = 0; m < MM; ++m) mx = fmaxf(mx, sc[m]);
      float sum = 0.0f;
      for (int m = 0; m < MM; ++m) sum += __expf(sc[m] - mx);
      red[0] = mx;
      red[1] = 1.0f / sum;
    }
    __syncthreads();
    if (tid < MM) sc[tid] = __expf(sc[tid] - red[0]) * red[1];
    __syncthreads();
    // u += sum_m p[m] * (ec[m] + TC[m])   (coalesced over d)
    if (tid < DD) {
      float o = 0.0f;
      for (int m = 0; m < MM; ++m)
        o += sc[m] * (ec[m * DD + tid] + TC[m * DD + tid]);
      u_s[tid] += o;
    }
    __syncthreads();
  }
  if (tid < DD) u[b * DD + tid] = u_s[tid];
}

// ---------------------------------------------------------------------------
// logits[64, 50000] = u[64,128] @ W[50000,128]^T + bias, via
// V_WMMA_F32_16X16X4_F32. Block = 256 threads = 8 waves; each wave owns a
// 16x16 tile: 4 M-blocks x 2 N-blocks. u staged in LDS (32KB).
// ---------------------------------------------------------------------------
__global__ void memn2n_logits(const float* __restrict__ u,
                              const float* __restrict__ W,
                              const float* __restrict__ bias,
                              float* __restrict__ out) {
  __shared__ float u_s[BB * DD];        // 32 KB
  for (int i = threadIdx.x; i < BB * DD; i += blockDim.x) u_s[i] = u[i];
  __syncthreads();

  const int l    = threadIdx.x & 31;
  const int w    = threadIdx.x >> 5;    // wave 0..7
  const int mb   = w & 3;               // M block (64 rows / 16)
  const int nb   = w >> 2;              // N sub-block 0..1
  const int arow = mb * 16 + (l & 15);  // A layout: M = lane&15 (both halves)
  const int kb   = (l >> 4) << 1;       // lane half K offset: 0 or 2
  const int j    = blockIdx.x * 32 + nb * 16 + (l & 15);
  const int jc   = (j < VV) ? j : (VV - 1);   // clamp; OOB cols never stored
  const float* wrow  = W + (size_t)jc * DD;
  const float* uarow = u_s + arow * DD;

  v8f acc = {};
#pragma unroll 8
  for (int k0 = 0; k0 < DD; k0 += 4) {
    v2f a  = *(const v2f*)(uarow + k0 + kb);  // A[m][k], consecutive K pair
    v2f bb = *(const v2f*)(wrow  + k0 + kb);  // B[k][n] = W[j][k]
    acc = __builtin_amdgcn_wmma_f32_16x16x4_f32(
        /*neg_a=*/false, a, /*neg_b=*/false, bb,
        /*c_mod=*/(short)0, acc, /*reuse_a=*/false, /*reuse_b=*/false);
  }

  if (j < VV) {
    const float bj = bias[j];
    const int hi = (l >> 4) * 8;        // C layout: VGPR r -> M = r + 8*(lane>=16)
#pragma unroll
    for (int r = 0; r < 8; ++r)
      out[(size_t)(mb * 16 + r + hi) * VV + j] = acc[r] + bj;
  }
}

// ---------------------------------------------------------------------------
extern "C" void kernel_launch(void* const* d_in, const int* in_sizes, int n_in,
                              void* d_out, int out_size, void* d_ws, size_t ws_size,
                              hipStream_t stream) {
  (void)in_sizes; (void)n_in; (void)out_size; (void)ws_size;
  const int*   story    = (const int*)d_in[0];
  const int*   question = (const int*)d_in[1];
  const float* A0   = (const float*)d_in[2];
  const float* C0   = (const float*)d_in[3];
  const float* C1   = (const float*)d_in[4];
  const float* C2   = (const float*)d_in[5];
  const float* Bemb = (const float*)d_in[6];
  const float* TA   = (const float*)d_in[7];
  const float* TC   = (const float*)d_in[8];
  const float* W    = (const float*)d_in[9];
  const float* bias = (const float*)d_in[10];
  float* out = (float*)d_out;

  float* ws  = (float*)d_ws;
  float* emb = ws;                                  // [4][B*M][D]  ~26.2 MB
  float* u   = ws + (size_t)4 * BB * MM * DD;       // [B][D]

  memn2n_embed<<<dim3(BB * MM, 4), 128, 0, stream>>>(story, A0, C0, C1, C2, emb);
  memn2n_query<<<BB, 128, 0, stream>>>(question, Bemb, u);
  memn2n_hops<<<BB, 256, 0, stream>>>(story, TA, TC, emb, u);
  memn2n_logits<<<(VV + 31) / 32, 256, 0, stream>>>(u, W, bias, out);
}